// MPSClassifier_58222576665171
// MI455X (gfx1250) — compile-verified
//
#include <hip/hip_runtime.h>

// MPS / tensor-train classifier on gfx1250 using V_WMMA_F32_16X16X4_F32.
// One wave32 = one 16-element batch tile. Bond=5 padded into the 16x16x4 tile:
//   M = output bond r (5/16 used), K = merged (l,i) index (10 -> 3 chunks of 4),
//   N = batch column (16/16 used).
// Per-batch feature scaling xi commutes as a column scaling of the B operand,
// so the shared core matrix stays a clean uniform A operand.
// Invalid A lanes (r>=5 / hi-half l=5) read a zeroed LDS pad region with
// per-lane stride 0 instead of being masked with v_cndmask each iteration.

typedef __attribute__((ext_vector_type(2))) float v2f;
typedef __attribute__((ext_vector_type(8))) float v8f;

#define B_TOT      16384
#define D_SITES    784
#define BOND       5
#define OUT_DIM    10
#define MID_SITES  (D_SITES - 2)                     // 782
#define CORE_FLOATS (MID_SITES * BOND * 2 * BOND)    // 39100 floats = 156.4 KB
#define ZPAD       256                               // covers unrolled DS imm offsets
#define LDS_FLOATS (CORE_FLOATS + ZPAD)
#define ZSLOT      CORE_FLOATS                       // start of zeroed pad (even idx)

__global__ __launch_bounds__(256) void mps_tt_wmma_kernel(
    const float* __restrict__ x,           // [B, D]
    const float* __restrict__ core_first,  // [1,2,BOND] -> cf[i*5 + l]
    const float* __restrict__ cores_mid,   // [782,BOND,2,BOND] -> [site][l][i][r]
    const float* __restrict__ core_last,   // [BOND,2,1]   -> cl[l*2 + i]
    const float* __restrict__ fc_w,        // [OUT,1]
    const float* __restrict__ fc_b,        // [OUT]
    float* __restrict__ out)               // [B, OUT]
{
    extern __shared__ float lds[];

    // Cooperative load of ALL mid cores into LDS (156.4KB < 320KB/WGP),
    // transposed to [site][l][r][i] so each lane's (i=0,i=1) pair is one b64.
    for (int j = threadIdx.x; j < CORE_FLOATS; j += blockDim.x) {
        int site = j / 50;
        int rem  = j - site * 50;
        int l    = rem / 10;
        int t    = rem - l * 10;
        int i    = t / 5;
        int r    = t - i * 5;
        lds[site * 50 + l * 10 + r * 2 + i] = cores_mid[j];
    }
    // Zero pad region: invalid lanes park their DS addresses here (stride 0),
    // so no per-iteration value masking is needed.
    for (int j = CORE_FLOATS + threadIdx.x; j < LDS_FLOATS; j += blockDim.x)
        lds[j] = 0.0f;
    __syncthreads();

    const int  lane = threadIdx.x & 31;
    const int  wave = threadIdx.x >> 5;
    const int  col  = lane & 15;              // batch column owned by this lane
    const bool hi   = lane >= 16;             // hi half supplies K=2,3 (l = 2c+1)
    const int  tile = blockIdx.x * (blockDim.x >> 5) + wave;
    const int  b    = tile * 16 + col;
    const float* xrow = x + (size_t)b * D_SITES;

    // ---------------- first core: carry0[l] = xs0*cf[0][l] + xs1*cf[1][l] ----
    float x0  = xrow[0];
    float xs0 = 1.0f - x0, xs1 = x0;
    float c0 = xs0 * core_first[0] + xs1 * core_first[5];
    float c1 = xs0 * core_first[1] + xs1 * core_first[6];
    float c2 = xs0 * core_first[2] + xs1 * core_first[7];
    float c3 = xs0 * core_first[3] + xs1 * core_first[8];
    float c4 = xs0 * core_first[4] + xs1 * core_first[9];
    // Carry rows resident per lane (B-operand layout):
    //   lo lane: rows 0,2,4 ; hi lane: rows 1,3
    float rowA = hi ? c1 : c0;
    float rowB = hi ? c3 : c2;
    float rowC = hi ? 0.0f : c4;

    const bool mvalid = (col < BOND);         // A rows r>=5 are zero padding
    const int  aoff   = hi ? 10 : 0;

    // Per-lane LDS cursors: invalid lanes sit on the zero pad with stride 0.
    int ai  = mvalid ? (aoff + col * 2) : ZSLOT;          // a0 @ +0, a1 @ +20
    int ai2 = (mvalid && !hi) ? (40 + col * 2) : ZSLOT;   // a2 (l=4, lo half only)
    const int as  = mvalid ? 50 : 0;
    const int as2 = (mvalid && !hi) ? 50 : 0;

    // ---------------- 782 sequential mid-site contractions -------------------
#pragma unroll 4
    for (int d = 1; d <= MID_SITES; ++d) {
        float xv  = xrow[d];
        float xi0 = 1.0f - xv, xi1 = xv;

        // B operand chunks: column scaling by (xi0, xi1) is per-lane math.
        v2f b0, b1, b2;
        b0.x = rowA * xi0;  b0.y = rowA * xi1;
        b1.x = rowB * xi0;  b1.y = rowB * xi1;
        b2.x = rowC * xi0;  b2.y = rowC * xi1;

        // A operand chunks from LDS [site][l][r][i]; lane m=col -> row r=m.
        // Invalid lanes read zeros from the pad region (no masking needed).
        v2f a0 = *(const v2f*)&lds[ai];
        v2f a1 = *(const v2f*)&lds[ai + 20];
        v2f a2 = *(const v2f*)&lds[ai2];
        ai  += as;
        ai2 += as2;

        // new_carry = A(core) x B(scaled carry), f32 end-to-end.
        v8f acc = {};
        acc = __builtin_amdgcn_wmma_f32_16x16x4_f32(false, a0, false, b0,
                                                    (short)0, acc, false, false);
        acc = __builtin_amdgcn_wmma_f32_16x16x4_f32(false, a1, false, b1,
                                                    (short)0, acc, false, false);
        acc = __builtin_amdgcn_wmma_f32_16x16x4_f32(false, a2, false, b2,
                                                    (short)0, acc, false, false);

        // C layout -> next B layout: lo lanes keep rows 0,2,4 in-lane;
        // hi lanes fetch rows 1,3 of their column from lane (lane&15).
        float s1 = __shfl(acc[1], col, 32);
        float s3 = __shfl(acc[3], col, 32);
        rowA = hi ? s1 : acc[0];
        rowB = hi ? s3 : acc[2];
        rowC = hi ? 0.0f : acc[4];
    }

    // ---------------- last core + FC head ------------------------------------
    float xl = xrow[D_SITES - 1];
    float y0 = 1.0f - xl, y1 = xl;
    float v0  = y0 * core_last[0] + y1 * core_last[1];
    float v1  = y0 * core_last[2] + y1 * core_last[3];
    float v2v = y0 * core_last[4] + y1 * core_last[5];
    float v3  = y0 * core_last[6] + y1 * core_last[7];
    float v4  = y0 * core_last[8] + y1 * core_last[9];

    float e0 = hi ? v1 : v0;
    float e1 = hi ? v3 : v2v;
    float e2 = hi ? 0.0f : v4;
    float partial = rowA * e0 + rowB * e1 + rowC * e2;

    // res[col] = lo-half partial + hi-half partial (broadcast to both halves)
    float res = __shfl(partial, col, 32) + __shfl(partial, col + 16, 32);

    // out[b, o] = res * fc_w[o] + fc_b[o]; lo half writes o=0..4, hi o=5..9.
    int obase = hi ? 5 : 0;
#pragma unroll
    for (int o = 0; o < 5; ++o) {
        int oo = obase + o;
        out[(size_t)b * OUT_DIM + oo] = res * fc_w[oo] + fc_b[oo];
    }
}

extern "C" void kernel_launch(void* const* d_in, const int* in_sizes, int n_in,
                              void* d_out, int out_size, void* d_ws, size_t ws_size,
                              hipStream_t stream) {
    const float* x  = (const float*)d_in[0];
    const float* cf = (const float*)d_in[1];
    const float* cm = (const float*)d_in[2];
    const float* cl = (const float*)d_in[3];
    const float* fw = (const float*)d_in[4];
    const float* fb = (const float*)d_in[5];
    float* out = (float*)d_out;

    dim3 block(256);                              // 8 waves / block
    dim3 grid(B_TOT / (16 * 8));                  // 1024 tiles -> 128 blocks
    size_t shmem = (size_t)LDS_FLOATS * sizeof(float);  // ~157.4 KB < 320 KB WGP LDS
    hipLaunchKernelGGL(mps_tt_wmma_kernel, grid, block, shmem, stream,
                       x, cf, cm, cl, fw, fb, out);
}